// HeteroGATReal_46136538693992
// MI455X (gfx1250) — compile-verified
//
#include <hip/hip_runtime.h>
#include <hip/hip_bf16.h>

typedef __attribute__((ext_vector_type(2))) float v2f;
typedef __attribute__((ext_vector_type(8))) float v8f;

#define HD   128   // H*D == IN == 128
#define NH   4     // heads
#define ALPHA 0.2f

// ---------------------------------------------------------------------------
// GEMM: Y[N,128] = X[N,128] @ W[128,128] + b, fp32 via V_WMMA_F32_16X16X4_F32.
// Block = 256 threads (8 waves), computes 128 rows. A tile (128x128 = 64KB)
// staged in LDS with coalesced float4 loads + XOR swizzle at 16B granularity
// (c4 ^ (row&31)) so per-lane fragment reads are bank-conflict free.
// A 16x4 f32 frag: lanes0-15 M=lane {K0,K1}; lanes16-31 M=lane-16 {K2,K3}.
// B 4x16 f32 frag (mirror): lanes0-15 N=lane {K0,K1}; lanes16-31 {K2,K3}.
// C/D 16x16 f32: vgpr r, lanes0-15 -> (M=r, N=lane); lanes16-31 -> (M=r+8).
// ---------------------------------------------------------------------------
__global__ __launch_bounds__(256) void hgat_gemm_bias_wmma(
    const float* __restrict__ X, const float* __restrict__ W,
    const float* __restrict__ bias, float* __restrict__ Y, int N) {
  __shared__ float lA[128 * HD];  // 64KB
  int bm0 = blockIdx.x * 128;
  int tid = threadIdx.x;

  // stage A: 4096 float4s, coalesced global reads, swizzled LDS writes
#pragma unroll
  for (int j = 0; j < 16; ++j) {
    int i = tid + 256 * j;  // 0..4095
    int r = i >> 5, c4 = i & 31;
    if (bm0 + r < N) {
      float4 v = *(const float4*)(X + (size_t)(bm0 + r) * HD + c4 * 4);
      *(float4*)(lA + r * HD + 4 * (c4 ^ (r & 31))) = v;
    }
  }
  __syncthreads();

  int wid = tid >> 5, lane = tid & 31;
  int m0 = bm0 + wid * 16;
  if (m0 >= N) return;  // wave-uniform exit after barrier (EXEC stays all-1s)
  int half = lane >> 4; // 0: K0/K1 ; 1: K2/K3
  int mrow = lane & 15;
  int lr = wid * 16 + mrow;

  // hoist A fragments from LDS once; reused across 8 column tiles
  v2f a[32];
#pragma unroll
  for (int s = 0; s < 32; ++s)
    a[s] = *(const v2f*)(lA + lr * HD + 4 * (s ^ (lr & 31)) + 2 * half);

#pragma unroll 1
  for (int nt = 0; nt < 8; ++nt) {
    int n0 = nt * 16;
    float bv = bias[n0 + mrow];
    v8f c;
#pragma unroll
    for (int r = 0; r < 8; ++r) c[r] = bv;
#pragma unroll
    for (int s = 0; s < 32; ++s) {
      const float* wp = W + (size_t)(4 * s + 2 * half) * HD + n0 + mrow;
      v2f b;
      b.x = wp[0];
      b.y = wp[HD];
      c = __builtin_amdgcn_wmma_f32_16x16x4_f32(false, a[s], false, b,
                                                (short)0, c, false, false);
    }
    float* yr = Y + (size_t)(m0 + 8 * half) * HD + n0 + mrow;
#pragma unroll
    for (int r = 0; r < 8; ++r) yr[(size_t)r * HD] = c[r];
  }
}

// ---------------------------------------------------------------------------
// Per-(node,head) attention projection: out[n*4+h] = dot(Wh[n,h,:], att[h,:])
// ---------------------------------------------------------------------------
__global__ void hgat_head_dot(const float* __restrict__ Wh,
                              const float* __restrict__ att,
                              float* __restrict__ out, int NH_total) {
  int i = blockIdx.x * blockDim.x + threadIdx.x;
  if (i >= NH_total) return;
  int h = i & 3, n = i >> 2;
  const float4* w = (const float4*)(Wh + (size_t)n * HD + h * 32);
  const float4* av = (const float4*)(att + h * 32);
  float acc = 0.f;
#pragma unroll
  for (int j = 0; j < 8; ++j) {
    float4 x = w[j], y = av[j];
    acc += x.x * y.x + x.y * y.y + x.z * y.z + x.w * y.w;
  }
  out[i] = acc;
}

__global__ void hgat_zero2(unsigned* __restrict__ m, float* __restrict__ s, int n) {
  int i = blockIdx.x * blockDim.x + threadIdx.x;
  if (i < n) { m[i] = 0u; s[i] = 0.f; }
}

__device__ __forceinline__ float leaky(float x) { return x > 0.f ? x : ALPHA * x; }
// monotone float <-> uint mapping for atomicMax
__device__ __forceinline__ unsigned f2ord(float f) {
  unsigned u = __float_as_uint(f);
  return (u & 0x80000000u) ? ~u : (u | 0x80000000u);
}
__device__ __forceinline__ float ord2f(unsigned k) {
  return __uint_as_float((k & 0x80000000u) ? (k & 0x7FFFFFFFu) : ~k);
}

// Pass A: segment max of leaky(es[src]+ed[dst]) keyed by dst
__global__ void hgat_edge_max(const int* __restrict__ src, const int* __restrict__ dst,
                              const float* __restrict__ es, const float* __restrict__ ed,
                              unsigned* __restrict__ m, int E) {
  int i = blockIdx.x * blockDim.x + threadIdx.x;
  if (i >= E) return;
  int si = src[i], di = dst[i];
  float4 s4 = *(const float4*)(es + (size_t)si * NH);
  float4 d4 = *(const float4*)(ed + (size_t)di * NH);
  unsigned* mb = m + (size_t)di * NH;
  atomicMax(mb + 0, f2ord(leaky(s4.x + d4.x)));
  atomicMax(mb + 1, f2ord(leaky(s4.y + d4.y)));
  atomicMax(mb + 2, f2ord(leaky(s4.z + d4.z)));
  atomicMax(mb + 3, f2ord(leaky(s4.w + d4.w)));
}

// decode keys -> float in place; empty segments (key 0 -> NaN) -> 0
__global__ void hgat_fin_max(unsigned* __restrict__ m, int n) {
  int i = blockIdx.x * blockDim.x + threadIdx.x;
  if (i >= n) return;
  float f = ord2f(m[i]);
  if ((__float_as_uint(f) & 0x7F800000u) == 0x7F800000u) f = 0.f;
  ((float*)m)[i] = f;
}

// Pass B: ee = exp(e - m[dst]); s[dst] += ee; cache ee per edge
__global__ void hgat_edge_exp(const int* __restrict__ src, const int* __restrict__ dst,
                              const float* __restrict__ es, const float* __restrict__ ed,
                              const float* __restrict__ m, float* __restrict__ s,
                              float* __restrict__ ee, int E) {
  int i = blockIdx.x * blockDim.x + threadIdx.x;
  if (i >= E) return;
  int si = src[i], di = dst[i];
  float4 s4 = *(const float4*)(es + (size_t)si * NH);
  float4 d4 = *(const float4*)(ed + (size_t)di * NH);
  float4 m4 = *(const float4*)(m + (size_t)di * NH);
  float4 e4;
  e4.x = __expf(leaky(s4.x + d4.x) - m4.x);
  e4.y = __expf(leaky(s4.y + d4.y) - m4.y);
  e4.z = __expf(leaky(s4.z + d4.z) - m4.z);
  e4.w = __expf(leaky(s4.w + d4.w) - m4.w);
  *(float4*)(ee + (size_t)i * NH) = e4;
  float* sb = s + (size_t)di * NH;
  atomicAdd(sb + 0, e4.x);
  atomicAdd(sb + 1, e4.y);
  atomicAdd(sb + 2, e4.z);
  atomicAdd(sb + 3, e4.w);
}

// Pass C: out[dst] += Wh_src[src] * (ee/s[dst]); 32 lanes per edge, float4/lane
__global__ void hgat_edge_scatter(const int* __restrict__ src, const int* __restrict__ dst,
                                  const float* __restrict__ WhS, const float* __restrict__ s,
                                  const float* __restrict__ ee, float* __restrict__ out, int E) {
  int t = blockIdx.x * blockDim.x + threadIdx.x;
  int edge = t >> 5, lane = t & 31;
  if (edge >= E) return;
  int si = src[edge], di = dst[edge];
  int h = lane >> 3;  // 8 lanes (32 floats) per head
  float a = ee[(size_t)edge * NH + h] / s[(size_t)di * NH + h];
  float4 w = *(const float4*)(WhS + (size_t)si * HD + lane * 4);
  float* o = out + (size_t)di * HD + lane * 4;
  atomicAdd(o + 0, w.x * a);
  atomicAdd(o + 1, w.y * a);
  atomicAdd(o + 2, w.z * a);
  atomicAdd(o + 3, w.w * a);
}

__global__ void hgat_relu(float* __restrict__ o, int n) {
  int i = blockIdx.x * blockDim.x + threadIdx.x;
  if (i < n) o[i] = fmaxf(o[i], 0.f);
}

static inline int cdiv(int a, int b) { return (a + b - 1) / b; }

extern "C" void kernel_launch(void* const* d_in, const int* in_sizes, int n_in,
                              void* d_out, int out_size, void* d_ws, size_t ws_size,
                              hipStream_t stream) {
  const float* feat_P = (const float*)d_in[0];
  const float* feat_A = (const float*)d_in[1];
  const int* srcI[4] = {(const int*)d_in[2], (const int*)d_in[4],
                        (const int*)d_in[6], (const int*)d_in[8]};
  const int* dstI[4] = {(const int*)d_in[3], (const int*)d_in[5],
                        (const int*)d_in[7], (const int*)d_in[9]};
  int Erel[4] = {in_sizes[2], in_sizes[4], in_sizes[6], in_sizes[8]};
  const float* W_P = (const float*)d_in[10];  const float* b_P = (const float*)d_in[11];
  const float* W_A = (const float*)d_in[12];  const float* b_A = (const float*)d_in[13];
  const float* Wrel[4] = {(const float*)d_in[14], (const float*)d_in[16],
                          (const float*)d_in[18], (const float*)d_in[20]};
  const float* brel[4] = {(const float*)d_in[15], (const float*)d_in[17],
                          (const float*)d_in[19], (const float*)d_in[21]};
  const float* attS[4] = {(const float*)d_in[22], (const float*)d_in[24],
                          (const float*)d_in[26], (const float*)d_in[28]};
  const float* attD[4] = {(const float*)d_in[23], (const float*)d_in[25],
                          (const float*)d_in[27], (const float*)d_in[29]};

  int N_P = in_sizes[0] / HD;
  int N_A = in_sizes[1] / HD;
  int NMAX = N_P > N_A ? N_P : N_A;
  int EMAX = Erel[0];
  for (int r = 1; r < 4; ++r) if (Erel[r] > EMAX) EMAX = Erel[r];

  float* outP = (float*)d_out;
  float* outA = outP + (size_t)N_P * HD;

  // workspace carve-up (all chunks 16B aligned)
  char* w = (char*)d_ws;
  float* wh = (float*)w;            w += (size_t)NMAX * HD * sizeof(float);
  float* esb = (float*)w;           w += (size_t)NMAX * NH * sizeof(float);
  float* edb[4];
  for (int r = 0; r < 4; ++r) { edb[r] = (float*)w; w += (size_t)NMAX * NH * sizeof(float); }
  unsigned* mb = (unsigned*)w;      w += (size_t)NMAX * NH * sizeof(unsigned);
  float* sb = (float*)w;            w += (size_t)NMAX * NH * sizeof(float);
  float* eeb = (float*)w;           w += (size_t)EMAX * NH * sizeof(float);

  // relation metadata: src feats, dst accumulator, sizes
  const float* srcFeat[4] = {feat_P, feat_P, feat_A, feat_A};
  float* outDst[4] = {outP, outA, outP, outA};
  int Ns[4] = {N_P, N_P, N_A, N_A};
  int Nd[4] = {N_P, N_A, N_P, N_A};

  // 1) self transforms straight into d_out
  hgat_gemm_bias_wmma<<<cdiv(N_P, 128), 256, 0, stream>>>(feat_P, W_P, b_P, outP, N_P);
  hgat_gemm_bias_wmma<<<cdiv(N_A, 128), 256, 0, stream>>>(feat_A, W_A, b_A, outA, N_A);

  // 2) all dst-side projections BEFORE any scatter mutates d_out
  {
    const float* whDst[4] = {outP, outA, outP, outA};
    for (int r = 0; r < 4; ++r)
      hgat_head_dot<<<cdiv(Nd[r] * NH, 256), 256, 0, stream>>>(whDst[r], attD[r], edb[r], Nd[r] * NH);
  }

  // 3) per-relation attention + aggregation
  for (int r = 0; r < 4; ++r) {
    int E = Erel[r];
    hgat_gemm_bias_wmma<<<cdiv(Ns[r], 128), 256, 0, stream>>>(
        srcFeat[r], Wrel[r], brel[r], wh, Ns[r]);
    hgat_head_dot<<<cdiv(Ns[r] * NH, 256), 256, 0, stream>>>(wh, attS[r], esb, Ns[r] * NH);
    hgat_zero2<<<cdiv(Nd[r] * NH, 256), 256, 0, stream>>>(mb, sb, Nd[r] * NH);
    hgat_edge_max<<<cdiv(E, 256), 256, 0, stream>>>(srcI[r], dstI[r], esb, edb[r], mb, E);
    hgat_fin_max<<<cdiv(Nd[r] * NH, 256), 256, 0, stream>>>(mb, Nd[r] * NH);
    hgat_edge_exp<<<cdiv(E, 256), 256, 0, stream>>>(srcI[r], dstI[r], esb, edb[r],
                                                    (const float*)mb, sb, eeb, E);
    hgat_edge_scatter<<<cdiv(E * 32, 256), 256, 0, stream>>>(srcI[r], dstI[r], wh, sb, eeb,
                                                             outDst[r], E);
  }

  // 4) final relu over both outputs
  hgat_relu<<<cdiv(out_size, 256), 256, 0, stream>>>((float*)d_out, out_size);
}